// ManualAdaptiveAvgPool2d_57483842290272
// MI455X (gfx1250) — compile-verified
//
#include <hip/hip_runtime.h>
#include <cstdint>

// Problem geometry (from reference): x = (64, 512, 52, 52) fp32,
// 4x4 sum-pool stride 4 -> (64, 512, 13, 13), out = clamp(round(sum/32), -128, 127)
#define PLANE_H      52
#define PLANE_W      52
#define OUT_H        13
#define OUT_W        13
#define KH           4
#define KW           4
#define PLANE_ELEMS  (PLANE_H * PLANE_W)   // 2704 floats = 10816 B per plane
#define OUT_ELEMS    (OUT_H * OUT_W)       // 169 outputs per plane
#define PLANES_PER_WG 4
#define BLOCK        256                   // 8 waves (wave32)
#define SCALE        0.03125f              // OUTPUT_SLICE/INPUT_SLICE/kh/kw = 1/32

__global__ __launch_bounds__(BLOCK)
void ManualAdaptiveAvgPool2d_pool_kernel(const float* __restrict__ x,
                                         float* __restrict__ out) {
    // 4 planes staged in LDS: 43,264 B (<< 320 KB/WGP; ~7 blocks resident per WGP)
    __shared__ __align__(16) float tile[PLANES_PER_WG * PLANE_ELEMS];

    const unsigned tid = threadIdx.x;
    const unsigned wg  = blockIdx.x;

    const float* src = x + (size_t)wg * (PLANES_PER_WG * PLANE_ELEMS);

    // LDS byte offset of tile[].
    const unsigned lds_base = (unsigned)(uintptr_t)(&tile[0]);

    // ---- Stage 4 planes global -> LDS with async B128 copies (ASYNCcnt path) ----
    const unsigned NCHUNK = (PLANES_PER_WG * PLANE_ELEMS) / 4;  // 2704 x 16B chunks
    for (unsigned k = tid; k < NCHUNK; k += BLOCK) {
        const float* g = src + 4u * k;          // 16B-aligned global source
        unsigned     l = lds_base + 16u * k;    // 16B-aligned LDS destination
        asm volatile("global_load_async_to_lds_b128 %0, %1, off"
                     :
                     : "v"(l), "v"(g)
                     : "memory");
    }

    // Each wave drains its own async copies, then the workgroup barrier makes
    // all staged data visible to every wave.
#if __has_builtin(__builtin_amdgcn_s_wait_asynccnt)
    __builtin_amdgcn_s_wait_asynccnt(0);
#else
    asm volatile("s_wait_asynccnt 0x0" ::: "memory");
#endif
    __syncthreads();

    // ---- Reduce: one 4x4 window per thread, 4x aligned ds_load_b128 each ----
    const unsigned NOUT = PLANES_PER_WG * OUT_ELEMS;  // 676
    for (unsigned o = tid; o < NOUT; o += BLOCK) {
        const unsigned p   = o / OUT_ELEMS;
        const unsigned rem = o - p * OUT_ELEMS;
        const unsigned oy  = rem / OUT_W;
        const unsigned ox  = rem - oy * OUT_W;

        const float* base = &tile[p * PLANE_ELEMS + (oy * KH) * PLANE_W + ox * KW];
        const float4 r0 = *(const float4*)(base);
        const float4 r1 = *(const float4*)(base + PLANE_W);
        const float4 r2 = *(const float4*)(base + 2 * PLANE_W);
        const float4 r3 = *(const float4*)(base + 3 * PLANE_W);

        const float s = (((r0.x + r0.y) + (r0.z + r0.w)) +
                         ((r1.x + r1.y) + (r1.z + r1.w))) +
                        (((r2.x + r2.y) + (r2.z + r2.w)) +
                         ((r3.x + r3.y) + (r3.z + r3.w)));

        float v = rintf(s * SCALE);              // v_rndne_f32: round-half-to-even
        v = fminf(fmaxf(v, -128.0f), 127.0f);    // clip after round, like reference

        // out index = (wg*4 + p)*169 + rem == wg*676 + o  (coalesced stores)
        out[(size_t)wg * NOUT + o] = v;
    }
}

extern "C" void kernel_launch(void* const* d_in, const int* in_sizes, int n_in,
                              void* d_out, int out_size, void* d_ws, size_t ws_size,
                              hipStream_t stream) {
    (void)n_in; (void)out_size; (void)d_ws; (void)ws_size;
    const float* x   = (const float*)d_in[0];
    float*       out = (float*)d_out;

    const int planes = in_sizes[0] / PLANE_ELEMS;       // 64*512 = 32768
    const int blocks = planes / PLANES_PER_WG;          // 8192 (divides exactly)

    ManualAdaptiveAvgPool2d_pool_kernel<<<blocks, BLOCK, 0, stream>>>(x, out);
}